// YOLOLoss_11398843203937
// MI455X (gfx1250) — compile-verified
//
#include <hip/hip_runtime.h>

// ---------------- problem constants (match reference) ----------------
#define S_GRID 26
#define NBOX   3
#define NCLS   80
#define CHBOX  (5 + NCLS)        // 85
#define TOTCH  (NBOX * CHBOX)    // 255
#define NCELL  (S_GRID * S_GRID) // 676
#define BATCHN 128
#define NTGT   50
#define LAMBDA_COORD 5.0f
#define LAMBDA_NOOBJ 0.5f

typedef __attribute__((ext_vector_type(2))) float v2f;
typedef __attribute__((ext_vector_type(8))) float v8f;

// Exact f32 32-lane reduction via V_WMMA_F32_16X16X4_F32.
// A(16x4): lane m<16 holds {acc_m, 0} (K=0,1); lane m+16 holds {acc_{m+16}, 0} (K=2,3).
// B = all-ones(4x16)  =>  D[m][n] = acc_m + acc_{m+16}  (same for every n).
// D layout: VGPR j, lanes 0-15 -> M=j; lanes 16-31 -> M=j+8.
// Sum of 8 D-VGPRs per lane = half reduction; xor-16 shuffle completes it.
__device__ __forceinline__ float wave_reduce_sum_wmma(float x) {
    v2f a; a[0] = x;    a[1] = 0.0f;
    v2f b; b[0] = 1.0f; b[1] = 1.0f;
    v8f c = {};
    c = __builtin_amdgcn_wmma_f32_16x16x4_f32(false, a, false, b,
                                              (short)0, c, false, false);
    float s = c[0] + c[1] + c[2] + c[3] + c[4] + c[5] + c[6] + c[7];
    s += __shfl_xor(s, 16, 32);   // combine lane halves -> full 32-lane sum
    return s;
}

// Block reduction (blockDim.x multiple of 32, <= 256). Result valid in thread 0.
__device__ __forceinline__ float block_reduce_sum(float x) {
    __shared__ float smem[8];
    float w = wave_reduce_sum_wmma(x);
    const int lane = threadIdx.x & 31;
    const int wid  = threadIdx.x >> 5;
    if (lane == 0) smem[wid] = w;
    __syncthreads();
    float r = 0.0f;
    if (threadIdx.x == 0) {
        const int nw = blockDim.x >> 5;
        for (int i = 0; i < nw; ++i) r += smem[i];
    }
    return r;
}

// ---------------- kernel A: zero the obj mask ----------------
__global__ void yolo_zero_mask(unsigned int* __restrict__ mask, int n) {
    int i = blockIdx.x * blockDim.x + threadIdx.x;
    if (i < n) mask[i] = 0u;
}

// ---------------- kernel B: per-target coord+class loss, scatter obj mask ----------------
// One thread per target (grid*block == BATCHN*NTGT exactly; no inactive lanes).
__global__ void yolo_target_loss(const float* __restrict__ pred,
                                 const float* __restrict__ tgt,
                                 unsigned int* __restrict__ mask,
                                 float* __restrict__ partials) {
    const int tid = blockIdx.x * blockDim.x + threadIdx.x;
    const int b = tid / NTGT;
    const int t = tid - b * NTGT;

    const float* tr = tgt + ((size_t)b * NTGT + t) * 5;
    const float cls_f = tr[0];
    const float cx = tr[1], cy = tr[2], w = tr[3], h = tr[4];

    const int gx = (int)(cx * (float)S_GRID);   // trunc == astype(int32)
    const int gy = (int)(cy * (float)S_GRID);
    const float valid = ((gx < S_GRID) && (gy < S_GRID)) ? 1.0f : 0.0f;
    const int gxc = min(max(gx, 0), S_GRID - 1);
    const int gyc = min(max(gy, 0), S_GRID - 1);

    // base: predictions[b, 0, gyc, gxc]; channel ch at +ch*NCELL
    const float* pc = pred + (size_t)b * TOTCH * NCELL + (size_t)gyc * S_GRID + gxc;

    const float p1 = pc[(size_t)1 * NCELL];
    const float p2 = pc[(size_t)2 * NCELL];
    const float p3 = pc[(size_t)3 * NCELL];
    const float p4 = pc[(size_t)4 * NCELL];
    float coord = (p1 - cx) * (p1 - cx) + (p2 - cy) * (p2 - cy)
                + (p3 - w) * (p3 - w) + (p4 - h) * (p4 - h);

    const int cls = (int)cls_f;   // trunc toward zero
    float clsloss = 0.0f;
#pragma unroll 4
    for (int c = 0; c < NCLS; ++c) {
        const float oh = (c == cls) ? 1.0f : 0.0f;
        const float d = pc[(size_t)(5 + c) * NCELL] - oh;
        clsloss = fmaf(d, d, clsloss);
    }

    const float loss = valid * (LAMBDA_COORD * coord + clsloss);

    // scatter obj mask (set-to-1 is idempotent; races benign; only when valid)
    if (valid != 0.0f) mask[b * NCELL + gyc * S_GRID + gxc] = 1u;

    const float r = block_reduce_sum(loss);   // uniform call: EXEC all-ones
    if (threadIdx.x == 0) partials[blockIdx.x] = r;
}

// ---------------- kernel C: sum (conf * noobj)^2 over all cells/boxes ----------------
// Coalesced: consecutive threads walk consecutive cells within a (b,nb) conf plane.
__global__ void yolo_noobj_loss(const float* __restrict__ pred,
                                const unsigned int* __restrict__ mask,
                                float* __restrict__ partials) {
    const int N = BATCHN * NBOX * NCELL;    // 259584
    const int stride = gridDim.x * blockDim.x;
    float acc = 0.0f;
    for (int i = blockIdx.x * blockDim.x + threadIdx.x; i < N; i += stride) {
        const int cell = i % NCELL;
        const int bn   = i / NCELL;         // b*NBOX + nb
        const int nb   = bn % NBOX;
        const int b    = bn / NBOX;
        const float conf =
            pred[((size_t)b * TOTCH + (size_t)nb * CHBOX) * NCELL + cell];
        float term = conf * conf;
        if (nb == 0 && mask[b * NCELL + cell] != 0u) term = 0.0f;  // noobj = 1 - obj
        acc += term;
    }
    const float r = block_reduce_sum(acc);  // loop reconverges; EXEC all-ones here
    if (threadIdx.x == 0) partials[blockIdx.x] = r;
}

// ---------------- kernel D: deterministic final combine ----------------
__global__ void yolo_finalize(const float* __restrict__ pB, int nB,
                              const float* __restrict__ pC, int nC,
                              float* __restrict__ out) {
    if (blockIdx.x == 0 && threadIdx.x == 0) {
        float sB = 0.0f, sC = 0.0f;
        for (int i = 0; i < nB; ++i) sB += pB[i];
        for (int i = 0; i < nC; ++i) sC += pC[i];
        out[0] = (sB + LAMBDA_NOOBJ * sC) / (float)BATCHN;
    }
}

extern "C" void kernel_launch(void* const* d_in, const int* in_sizes, int n_in,
                              void* d_out, int out_size, void* d_ws, size_t ws_size,
                              hipStream_t stream) {
    (void)in_sizes; (void)n_in; (void)out_size; (void)ws_size;

    const float* pred = (const float*)d_in[0];   // [128, 255, 26, 26] f32
    const float* tgt  = (const float*)d_in[1];   // [128, 50, 5] f32
    float* out = (float*)d_out;                  // scalar f32

    // workspace layout
    char* ws = (char*)d_ws;
    const int maskN = BATCHN * NCELL;                            // 86528
    unsigned int* mask = (unsigned int*)ws;                      // 346112 B
    const int gridB = (BATCHN * NTGT) / 256;                     // 25 (exact)
    const int gridC = 512;                                       // grid-stride
    float* partialsB = (float*)(ws + (size_t)maskN * sizeof(unsigned int));
    float* partialsC = (float*)(ws + (size_t)maskN * sizeof(unsigned int) + 256);

    yolo_zero_mask<<<(maskN + 255) / 256, 256, 0, stream>>>(mask, maskN);
    yolo_target_loss<<<gridB, 256, 0, stream>>>(pred, tgt, mask, partialsB);
    yolo_noobj_loss<<<gridC, 256, 0, stream>>>(pred, mask, partialsC);
    yolo_finalize<<<1, 32, 0, stream>>>(partialsB, gridB, partialsC, gridC, out);
}